// PopulationODE_32255204393035
// MI455X (gfx1250) — compile-verified
//
#include <hip/hip_runtime.h>
#include <hip/hip_bf16.h>

typedef float v2f __attribute__((ext_vector_type(2)));

#define MAXL 512
#define TPB  256
#define VEC  2

#if defined(__HIP_DEVICE_COMPILE__) && \
    __has_builtin(__builtin_amdgcn_global_load_async_to_lds_b32) && \
    __has_builtin(__builtin_amdgcn_s_wait_asynccnt)
#define HAVE_ASYNC_LDS 1
#else
#define HAVE_ASYNC_LDS 0
#endif

typedef __attribute__((address_space(1))) int gint_t;   // global int
typedef __attribute__((address_space(3))) int lint_t;   // LDS int

__device__ __forceinline__ v2f splat2(float a) { v2f r = {a, a}; return r; }
// packed fma: d = a*b + c  (targets v_pk_fma_f32)
__device__ __forceinline__ v2f fma2(float a, v2f b, v2f c) {
    return __builtin_elementwise_fma(splat2(a), b, c);
}

__global__ __launch_bounds__(TPB)
void popode_euler_kernel(const float* __restrict__ s_grid,
                         const float* __restrict__ y0,
                         const float* __restrict__ w,
                         float* __restrict__ out,
                         int Lr, int Br)
{
    __shared__ float s_sh[MAXL];
    __shared__ float h_sh[MAXL];

    const int tid = threadIdx.x;

    // ---- Stage s_grid into LDS (CDNA5 async global->LDS path when available) ----
#if HAVE_ASYNC_LDS
    for (int i = tid; i < Lr; i += TPB) {
        __builtin_amdgcn_global_load_async_to_lds_b32(
            (gint_t*)(s_grid + i),
            (lint_t*)(s_sh + i),
            /*offset=*/0, /*cpol=*/0);
    }
    __builtin_amdgcn_s_wait_asynccnt(0);
#else
    for (int i = tid; i < Lr; i += TPB) s_sh[i] = s_grid[i];
#endif
    __syncthreads();

    // h_sh[l] is the step size used to produce output row l (l >= 1)
    for (int i = tid; i < Lr; i += TPB)
        h_sh[i] = (i >= 1) ? (s_sh[i] - s_sh[i - 1]) : 0.0f;
    __syncthreads();

    const int gid = blockIdx.x * TPB + tid;
    const int b0  = gid * VEC;           // base batch index (2 trajectories/thread)
    if (b0 + VEC > Br) return;

    // Uniform weights -> scalar loads (SGPR broadcast into VOP3P sources)
    float wr[21];
#pragma unroll
    for (int k = 0; k < 21; ++k) wr[k] = w[k];

    // Register-resident state, 2 trajectories per thread
    v2f A = *(const v2f*)(y0 + 0 * Br + b0);
    v2f T = *(const v2f*)(y0 + 1 * Br + b0);
    v2f N = *(const v2f*)(y0 + 2 * Br + b0);
    v2f C = *(const v2f*)(y0 + 3 * Br + b0);

    const size_t rowStride = (size_t)4 * (size_t)Br;   // floats per output row

    // Row 0 = y0 (streaming stores; output never re-read by this kernel)
    {
        float* o = out + b0;
        __builtin_nontemporal_store(A, (v2f*)(o + 0 * Br));
        __builtin_nontemporal_store(T, (v2f*)(o + 1 * Br));
        __builtin_nontemporal_store(N, (v2f*)(o + 2 * Br));
        __builtin_nontemporal_store(C, (v2f*)(o + 3 * Br));
    }

#pragma unroll 2
    for (int l = 1; l < Lr; ++l) {
        const float h = h_sh[l];                 // broadcast ds_load
        const v2f  hh = splat2(h);

        // All derivatives from the *current* state (simultaneous RHS)
        const v2f A2 = A * A, T2 = T * T, N2 = N * N, C2 = C * C;
        const v2f AT = A * T, TN = T * N, NC = N * C;

        const v2f dA = fma2(wr[2], A2, fma2(wr[1], A, splat2(wr[0])));
        const v2f dT = fma2(wr[8], AT,
                       fma2(wr[7], A2,
                       fma2(wr[6], A,
                       fma2(wr[5], T2,
                       fma2(wr[4], T, splat2(wr[3]))))));
        const v2f dN = fma2(wr[14], TN,
                       fma2(wr[13], T2,
                       fma2(wr[12], T,
                       fma2(wr[11], N2,
                       fma2(wr[10], N, splat2(wr[9]))))));
        const v2f dC = fma2(wr[20], NC,
                       fma2(wr[19], N2,
                       fma2(wr[18], N,
                       fma2(wr[17], C2,
                       fma2(wr[16], C, splat2(wr[15]))))));

        // Explicit Euler update
        A = __builtin_elementwise_fma(hh, dA, A);
        T = __builtin_elementwise_fma(hh, dT, T);
        N = __builtin_elementwise_fma(hh, dN, N);
        C = __builtin_elementwise_fma(hh, dC, C);

        float* o = out + (size_t)l * rowStride + b0;
        __builtin_nontemporal_store(A, (v2f*)(o + 0 * Br));
        __builtin_nontemporal_store(T, (v2f*)(o + 1 * Br));
        __builtin_nontemporal_store(N, (v2f*)(o + 2 * Br));
        __builtin_nontemporal_store(C, (v2f*)(o + 3 * Br));
    }
}

extern "C" void kernel_launch(void* const* d_in, const int* in_sizes, int n_in,
                              void* d_out, int out_size, void* d_ws, size_t ws_size,
                              hipStream_t stream) {
    const float* s_grid = (const float*)d_in[0];   // [L]
    const float* y0     = (const float*)d_in[1];   // [4, B]
    const float* w      = (const float*)d_in[2];   // [21]
    float* out = (float*)d_out;                    // [L, 4, B]

    const int L = in_sizes[0];
    const int B = in_sizes[1] / 4;

    const int threads = B / VEC;                   // 2 trajectories per thread
    const int blocks  = (threads + TPB - 1) / TPB;

    popode_euler_kernel<<<blocks, TPB, 0, stream>>>(s_grid, y0, w, out, L, B);
}